// bandWiseConv_1425929142665
// MI455X (gfx1250) — compile-verified
//
#include <hip/hip_runtime.h>
#include <math.h>

// Problem geometry (fixed by the reference): x is (8, 8, 64, 128, 128) f32.
#define CCH   64
#define HH    128
#define WW    128
#define TH    32            // output rows per block
#define TROWS (TH + 2)      // 34 LDS rows (with top/bottom halo)
// LDS row stride in floats. 136*4 = 544 B: every row start and every 4-float
// interior segment is 16B aligned. Columns: 3 = left zero halo,
// 4..131 = input cols 0..127, 132 = right zero halo.
#define TSTR  136

#if defined(__has_builtin)
#  if __has_builtin(__builtin_amdgcn_tensor_load_to_lds)
#    define USE_TDM 1
#  endif
#endif

typedef unsigned int u32x4 __attribute__((ext_vector_type(4)));
typedef int          i32x4 __attribute__((ext_vector_type(4)));
typedef int          i32x8 __attribute__((ext_vector_type(8)));
typedef float        f32x4 __attribute__((ext_vector_type(4)));

__global__ __launch_bounds__(256) void dwconv3x3_gelu_bn(
    const float* __restrict__ x,     const float* __restrict__ cw,
    const float* __restrict__ cb,    const float* __restrict__ gamma,
    const float* __restrict__ beta,  const float* __restrict__ rmean,
    const float* __restrict__ rvar,  float* __restrict__ out)
{
  __shared__ __align__(16) float tile[TROWS * TSTR];

  const int tid   = threadIdx.x;
  const int plane = blockIdx.y;            // 0..4095 == bg*64 + ch
  const int ch    = plane & (CCH - 1);
  const int row0  = blockIdx.x * TH;

  const float* src = x + (size_t)plane * (HH * WW);

  // ---- zero the left/right halo columns (DS stores, DScnt) ----
  for (int i = tid; i < TROWS * 2; i += 256) {
    int r = i >> 1;
    tile[r * TSTR + ((i & 1) ? (WW + 4) : 3)] = 0.0f;
  }

#ifdef USE_TDM
  // ================= Tensor Data Mover fill =================
  // One TDM op copies the whole (nrows x 128) f32 tile, row stride 128 elems,
  // into LDS with 8-DWORD padding after every 128 DWORDs (-> 136-float rows).
  // Bottom halo row comes from TDM OOB-reads-return-zero; top halo (row -1)
  // is produced by shifting the LDS start one row down and zeroing row 0.
  if (tid < 32) {          // exactly one wave issues the DMA
    const int grow0 = (row0 == 0) ? 0 : row0 - 1;        // first global row
    const int nrows = (row0 == 0) ? (TROWS - 1) : TROWS; // tile_dim1
    const int remro = HH - grow0;                        // rows to tensor end
    const int tr0   = (row0 == 0) ? 1 : 0;               // first LDS tile row
    const unsigned lds0 = (unsigned)(size_t)&tile[tr0 * TSTR + 4];
    const unsigned long long ga =
        (unsigned long long)(size_t)(src + (size_t)grow0 * WW);

    u32x4 g0;
    g0[0] = 1u;                                       // count=1, user mode
    g0[1] = lds0;                                     // lds_addr (bytes)
    g0[2] = (unsigned)(ga & 0xffffffffull);           // global_addr[31:0]
    g0[3] = (unsigned)((ga >> 32) & 0x01ffffffull)    // global_addr[56:32]
          | 0x80000000u;                              // type=2 ("image")

    i32x8 g1;
    g1[0] = (int)((2u << 16)      // data_size = 4B
                | (1u << 20)      // pad_enable
                | (6u << 22)      // pad_interval: 128 DWORDs
                | (7u << 25));    // pad_amount:   8 DWORDs
    g1[1] = (int)((unsigned)WW << 16);                 // tensor_dim0 = 128
    g1[2] = (int)(((unsigned)remro & 0xffffu) << 16);  // tensor_dim1 (OOB)
    g1[3] = (int)((unsigned)WW << 16);                 // tile_dim0 = 128
    g1[4] = nrows;                                     // tile_dim1 (tile_dim2=0)
    g1[5] = WW;                                        // tensor_dim0_stride
    g1[6] = 0;
    g1[7] = 0;

    i32x4 gz = {0, 0, 0, 0};
#if __has_include(<hip/amd_detail/amd_gfx1250_TDM.h>)
    i32x8 gz8 = {0, 0, 0, 0, 0, 0, 0, 0};
    __builtin_amdgcn_tensor_load_to_lds(g0, g1, gz, gz, gz8, 0);  // clang-23 form
#else
    __builtin_amdgcn_tensor_load_to_lds(g0, g1, gz, gz, 0);       // ROCm 7.2 form
#endif
  }
  if (row0 == 0) {         // zero the top halo row interior (SAME padding)
    for (int i = tid; i < WW; i += 256) tile[4 + i] = 0.0f;
  }
#if __has_builtin(__builtin_amdgcn_s_wait_tensorcnt)
  __builtin_amdgcn_s_wait_tensorcnt(0);
#else
  asm volatile("s_wait_tensorcnt 0x0" ::: "memory");
#endif
  __syncthreads();
#else
  // ============ fallback: per-lane async global->LDS fill ============
  const int nslots = TROWS * (WW / 4);     // 1088 float4 segments
  for (int s = tid; s < nslots; s += 256) {
    const int r   = s >> 5;                // 0..33   (tile row)
    const int seg = s & 31;                // 0..31   (4-float segment)
    const int gr  = row0 - 1 + r;          // global input row
    float* lp = &tile[r * TSTR + 4 + seg * 4];
    if (gr >= 0 && gr < HH) {
      unsigned lds_off = (unsigned)(size_t)lp;   // low 32 bits = LDS offset
      const float* g = src + gr * WW + seg * 4;
      asm volatile("global_load_async_to_lds_b128 %0, %1, off"
                   :: "v"(lds_off), "v"(g) : "memory");
    } else {
      f32x4 z = {0.f, 0.f, 0.f, 0.f};
      *(f32x4*)lp = z;
    }
  }
#if __has_builtin(__builtin_amdgcn_s_wait_asynccnt)
  __builtin_amdgcn_s_wait_asynccnt(0);
#else
  asm volatile("s_wait_asynccnt 0" ::: "memory");
#endif
  __syncthreads();
#endif

  // ---- per-channel constants (uniform across block -> scalar loads) ----
  const float w0 = cw[ch * 9 + 0], w1 = cw[ch * 9 + 1], w2 = cw[ch * 9 + 2];
  const float w3 = cw[ch * 9 + 3], w4 = cw[ch * 9 + 4], w5 = cw[ch * 9 + 5];
  const float w6 = cw[ch * 9 + 6], w7 = cw[ch * 9 + 7], w8 = cw[ch * 9 + 8];
  const float bias  = cb[ch];
  const float inv   = rsqrtf(rvar[ch] + 1e-5f);
  const float scale = gamma[ch] * inv;
  const float shift = beta[ch] - rmean[ch] * scale;

  // ---- compute: each thread owns a 4x4 output patch; sliding 3-row window ----
  const int cq  = tid & 31;        // column quad -> output cols cq*4 .. cq*4+3
  const int rg  = tid >> 5;        // row group   -> local rows  rg*4 .. rg*4+3
  const int c0  = cq * 4;
  const int lr0 = rg * 4;

  float t[6], m[6], b[6];          // 6-wide windows of 3 consecutive input rows
#pragma unroll
  for (int k = 0; k < 6; ++k) t[k] = tile[(lr0 + 0) * TSTR + 3 + c0 + k];
#pragma unroll
  for (int k = 0; k < 6; ++k) m[k] = tile[(lr0 + 1) * TSTR + 3 + c0 + k];

  float* dst = out + (size_t)plane * (HH * WW);
#pragma unroll
  for (int i = 0; i < 4; ++i) {
#pragma unroll
    for (int k = 0; k < 6; ++k) b[k] = tile[(lr0 + 2 + i) * TSTR + 3 + c0 + k];

    f32x4 o4;
#pragma unroll
    for (int j = 0; j < 4; ++j) {
      float acc = bias;
      acc += w0 * t[j] + w1 * t[j + 1] + w2 * t[j + 2];
      acc += w3 * m[j] + w4 * m[j + 1] + w5 * m[j + 2];
      acc += w6 * b[j] + w7 * b[j + 1] + w8 * b[j + 2];
      // exact-erf GELU (matches jax.nn.gelu(approximate=False))
      float g = 0.5f * acc * (1.0f + erff(acc * 0.70710678118654752f));
      o4[j] = g * scale + shift;
    }
    // streaming store: output is never re-read -> keep it out of L2's way
    __builtin_nontemporal_store(o4,
        (f32x4*)&dst[(size_t)(row0 + lr0 + i) * WW + c0]);

#pragma unroll
    for (int k = 0; k < 6; ++k) { t[k] = m[k]; m[k] = b[k]; }
  }
}

extern "C" void kernel_launch(void* const* d_in, const int* in_sizes, int n_in,
                              void* d_out, int out_size, void* d_ws, size_t ws_size,
                              hipStream_t stream) {
  const float* x     = (const float*)d_in[0];
  const float* cw    = (const float*)d_in[1];
  const float* cb    = (const float*)d_in[2];
  const float* gamma = (const float*)d_in[3];
  const float* beta  = (const float*)d_in[4];
  const float* rmean = (const float*)d_in[5];
  const float* rvar  = (const float*)d_in[6];
  float* out = (float*)d_out;

  const int planes = in_sizes[0] / (HH * WW);   // 8*8*64 = 4096
  dim3 grid(HH / TH, planes);                   // (4, 4096)
  dwconv3x3_gelu_bn<<<grid, 256, 0, stream>>>(x, cw, cb, gamma, beta,
                                              rmean, rvar, out);
}